// GraphAttentionV2Layer_41558103556501
// MI455X (gfx1250) — compile-verified
//
#include <hip/hip_runtime.h>

typedef __attribute__((ext_vector_type(2))) float v2f;
typedef __attribute__((ext_vector_type(8))) float v8f;

#define NNODES 100000
#define NEDGES 800000
#define NEG_SLOPE 0.2f
#define LN_EPS 1e-5f

// ---------------------------------------------------------------- helpers

__device__ __forceinline__ float wave_reduce_sum(float v) {
#pragma unroll
  for (int m = 16; m >= 1; m >>= 1) v += __shfl_xor(v, m, 32);
  return v;
}

__device__ __forceinline__ float wave_reduce_max(float v) {
#pragma unroll
  for (int m = 16; m >= 1; m >>= 1) v = fmaxf(v, __shfl_xor(v, m, 32));
  return v;
}

__device__ __forceinline__ int lower_bound_i32(const int* __restrict__ a, int n, int key) {
  int lo = 0, hi = n;
  while (lo < hi) {
    int mid = (lo + hi) >> 1;
    if (a[mid] < key) lo = mid + 1; else hi = mid;
  }
  return lo;
}

// B fragment for C = X * W^T : B[k][j] = W[j][k], W row-major [Dout, K].
// 32-bit 16x4 operand layout: lanes 0-15 hold (row=lane, k, k+1),
// lanes 16-31 hold (row=lane-16, k+2, k+3).
__device__ __forceinline__ v2f frag_rowmajor(const float* __restrict__ p,
                                             int row_base, int ld, int k, int lane) {
  int row = row_base + (lane & 15);
  int kk  = k + ((lane >> 4) << 1);
  const float* q = p + (size_t)row * ld + kk;
  v2f r; r.x = q[0]; r.y = q[1];
  return r;
}

// ---------------------------------------------------------------- kernel 1
// Y[R,64] = X[R,64] @ W[64,64]^T.  16 rows per block; 4 waves, 16 cols each.
__global__ __launch_bounds__(128) void proj64_kernel(const float* __restrict__ X,
                                                     const float* __restrict__ W,
                                                     float* __restrict__ Y) {
  __shared__ float As[16 * 66];   // +2 pad: conflict-free column reads
  const int tid  = threadIdx.x;
  const int lane = tid & 31;
  const int wid  = tid >> 5;
  const int m0   = blockIdx.x << 4;

  for (int i = tid; i < 16 * 64; i += 128) {
    int r = i >> 6, c = i & 63;
    As[r * 66 + c] = X[(size_t)(m0 + r) * 64 + c];
  }
  __syncthreads();

  const int j0 = wid << 4;
  v8f acc = {};
#pragma unroll
  for (int k = 0; k < 64; k += 4) {
    int arow = lane & 15;
    int akk  = k + ((lane >> 4) << 1);
    v2f a; a.x = As[arow * 66 + akk]; a.y = As[arow * 66 + akk + 1];
    v2f b = frag_rowmajor(W, j0, 64, k, lane);
    acc = __builtin_amdgcn_wmma_f32_16x16x4_f32(false, a, false, b, (short)0, acc,
                                                false, false);
  }

  const int col   = j0 + (lane & 15);
  const int rbase = m0 + ((lane >> 4) << 3);
#pragma unroll
  for (int v = 0; v < 8; ++v)
    Y[(size_t)(rbase + v) * 64 + col] = acc[v];
}

// ---------------------------------------------------------------- kernel 2
// Fused: g_r tile = e @ Wr^T (WMMA) -> global + LDS, then edge scores
// score[e] = sum_j leaky(concat(g_l[recv[e]], g_r[e]))[j] * wa[j]
// 16 edges per block; saves a full 205MB re-read of g_r.
__global__ __launch_bounds__(128) void proj_edge_score_kernel(
    const float* __restrict__ X,      // e [E,64]
    const float* __restrict__ W,      // Wr [64,64]
    const int*   __restrict__ recv,   // [E]
    const float* __restrict__ wa,     // [128]
    const float* __restrict__ gl,     // [N,64]
    float* __restrict__ Y,            // g_r out [E,64]
    float* __restrict__ score) {      // [E]
  __shared__ float As[16 * 66];
  __shared__ float Gs[16 * 66];       // g_r tile for score pass
  const int tid  = threadIdx.x;
  const int lane = tid & 31;
  const int wid  = tid >> 5;
  const int m0   = blockIdx.x << 4;

  for (int i = tid; i < 16 * 64; i += 128) {
    int r = i >> 6, c = i & 63;
    As[r * 66 + c] = X[(size_t)(m0 + r) * 64 + c];
  }
  __syncthreads();

  const int j0 = wid << 4;
  v8f acc = {};
#pragma unroll
  for (int k = 0; k < 64; k += 4) {
    int arow = lane & 15;
    int akk  = k + ((lane >> 4) << 1);
    v2f a; a.x = As[arow * 66 + akk]; a.y = As[arow * 66 + akk + 1];
    v2f b = frag_rowmajor(W, j0, 64, k, lane);
    acc = __builtin_amdgcn_wmma_f32_16x16x4_f32(false, a, false, b, (short)0, acc,
                                                false, false);
  }

  const int col = j0 + (lane & 15);
  const int rb  = (lane >> 4) << 3;
#pragma unroll
  for (int v = 0; v < 8; ++v) {
    Y[(size_t)(m0 + rb + v) * 64 + col] = acc[v];
    Gs[(rb + v) * 66 + col] = acc[v];
  }
  __syncthreads();

  // Score pass: each wave owns 4 edges of the 16-edge tile.
  const int d0 = lane << 1;
  v2f w0 = *(const v2f*)(wa + d0);
  v2f w1 = *(const v2f*)(wa + 64 + d0);
#pragma unroll
  for (int q = 0; q < 4; ++q) {
    const int et = (wid << 2) + q;          // edge within tile
    const int e  = m0 + et;                 // global edge id
    const int r  = recv[e];
    v2f a = *(const v2f*)(gl + (size_t)r * 64 + d0);    // gathered, L2-resident
    v2f b; b.x = Gs[et * 66 + d0]; b.y = Gs[et * 66 + d0 + 1];

    float ax = a.x >= 0.f ? a.x : NEG_SLOPE * a.x;
    float ay = a.y >= 0.f ? a.y : NEG_SLOPE * a.y;
    float bx = b.x >= 0.f ? b.x : NEG_SLOPE * b.x;
    float by = b.y >= 0.f ? b.y : NEG_SLOPE * b.y;

    float p = ax * w0.x + ay * w0.y + bx * w1.x + by * w1.y;
    p = wave_reduce_sum(p);
    if (lane == 0) score[e] = p;
  }
}

// ---------------------------------------------------------------- kernel 3
// One wave per node: segment softmax over sorted receivers + weighted aggregation.
// Receivers sorted => each node's edges are a contiguous range => no atomics.
__global__ __launch_bounds__(256) void seg_softmax_agg_kernel(const float* __restrict__ score,
                                                              const float* __restrict__ gr,
                                                              const int* __restrict__ recv,
                                                              float* __restrict__ agg) {
  const int lane = threadIdx.x & 31;
  const int node = (blockIdx.x << 3) + (threadIdx.x >> 5);

  const int lo = lower_bound_i32(recv, NEDGES, node);
  const int hi = lower_bound_i32(recv, NEDGES, node + 1);

  float m = -__builtin_inff();
  for (int i = lo + lane; i < hi; i += 32) m = fmaxf(m, score[i]);
  m = wave_reduce_max(m);
  if (lo >= hi) m = 0.f;   // reference: non-finite seg_max -> 0

  float s = 0.f;
  for (int i = lo + lane; i < hi; i += 32) s += expf(score[i] - m);
  s = wave_reduce_sum(s);
  const float inv = (s > 0.f) ? 1.f / s : 0.f;

  const int d0 = lane << 1;
  float acc0 = 0.f, acc1 = 0.f;
  for (int i = lo; i < hi; ++i) {
    float w = expf(score[i] - m) * inv;
    v2f g = *(const v2f*)(gr + (size_t)i * 64 + d0);   // sequential stream
    acc0 += w * g.x;
    acc1 += w * g.y;
  }
  v2f o; o.x = acc0; o.y = acc1;
  *(v2f*)(agg + (size_t)node * 64 + d0) = o;
}

// ---------------------------------------------------------------- kernel 4
// Fused node MLP + LayerNorm: x=[g_l|agg|u] (16x192) -> relu(x W1^T+b1) -> W2^T+b2 -> LN
__global__ __launch_bounds__(128) void mlp_ln_kernel(const float* __restrict__ gl,
                                                     const float* __restrict__ agg,
                                                     const float* __restrict__ u,
                                                     const float* __restrict__ W1,
                                                     const float* __restrict__ b1,
                                                     const float* __restrict__ W2,
                                                     const float* __restrict__ b2,
                                                     const float* __restrict__ gamma,
                                                     const float* __restrict__ beta,
                                                     float* __restrict__ out) {
  __shared__ float xs[16 * 194];
  __shared__ float x1s[16 * 66];
  __shared__ float x2s[16 * 66];
  __shared__ float mu_s[16], rs_s[16];

  const int tid  = threadIdx.x;
  const int lane = tid & 31;
  const int wid  = tid >> 5;
  const int m0   = blockIdx.x << 4;

  for (int i = tid; i < 16 * 192; i += 128) {
    int r = i / 192, c = i % 192;
    size_t row = (size_t)(m0 + r) * 64;
    float v;
    if (c < 64)       v = gl [row + c];
    else if (c < 128) v = agg[row + c - 64];
    else              v = u  [row + c - 128];
    xs[r * 194 + c] = v;
  }
  __syncthreads();

  const int j0 = wid << 4;

  // x1 = relu(x @ W1^T + b1), K = 192
  v8f acc = {};
#pragma unroll
  for (int k = 0; k < 192; k += 4) {
    int arow = lane & 15;
    int akk  = k + ((lane >> 4) << 1);
    v2f a; a.x = xs[arow * 194 + akk]; a.y = xs[arow * 194 + akk + 1];
    v2f b = frag_rowmajor(W1, j0, 192, k, lane);
    acc = __builtin_amdgcn_wmma_f32_16x16x4_f32(false, a, false, b, (short)0, acc,
                                                false, false);
  }
  {
    const int col  = j0 + (lane & 15);
    const float bb = b1[col];
    const int rb   = (lane >> 4) << 3;
#pragma unroll
    for (int v = 0; v < 8; ++v) {
      float x1 = acc[v] + bb;
      x1s[(rb + v) * 66 + col] = x1 > 0.f ? x1 : 0.f;
    }
  }
  __syncthreads();

  // x2 = x1 @ W2^T + b2, K = 64
  v8f acc2 = {};
#pragma unroll
  for (int k = 0; k < 64; k += 4) {
    int arow = lane & 15;
    int akk  = k + ((lane >> 4) << 1);
    v2f a; a.x = x1s[arow * 66 + akk]; a.y = x1s[arow * 66 + akk + 1];
    v2f b = frag_rowmajor(W2, j0, 64, k, lane);
    acc2 = __builtin_amdgcn_wmma_f32_16x16x4_f32(false, a, false, b, (short)0, acc2,
                                                 false, false);
  }
  {
    const int col  = j0 + (lane & 15);
    const float bb = b2[col];
    const int rb   = (lane >> 4) << 3;
#pragma unroll
    for (int v = 0; v < 8; ++v)
      x2s[(rb + v) * 66 + col] = acc2[v] + bb;
  }
  __syncthreads();

  // LayerNorm over 64 features per row
  if (tid < 16) {
    float s = 0.f, s2 = 0.f;
    for (int c = 0; c < 64; ++c) {
      float x = x2s[tid * 66 + c];
      s += x; s2 += x * x;
    }
    float mu  = s * (1.f / 64.f);
    float var = s2 * (1.f / 64.f) - mu * mu;
    mu_s[tid] = mu;
    rs_s[tid] = rsqrtf(var + LN_EPS);
  }
  __syncthreads();
  {
    int r  = tid >> 3;
    int c0 = (tid & 7) << 3;
    float mu = mu_s[r], rs = rs_s[r];
#pragma unroll
    for (int j = 0; j < 8; ++j) {
      int c = c0 + j;
      out[(size_t)(m0 + r) * 64 + c] = gamma[c] * (x2s[r * 66 + c] - mu) * rs + beta[c];
    }
  }
}

// ---------------------------------------------------------------- launch

extern "C" void kernel_launch(void* const* d_in, const int* in_sizes, int n_in,
                              void* d_out, int out_size, void* d_ws, size_t ws_size,
                              hipStream_t stream) {
  const float* h     = (const float*)d_in[0];
  const float* e     = (const float*)d_in[1];
  const int*   recv  = (const int*)  d_in[2];
  const float* u     = (const float*)d_in[3];
  const float* Wl    = (const float*)d_in[4];
  const float* Wr    = (const float*)d_in[5];
  const float* wa    = (const float*)d_in[6];
  const float* W1    = (const float*)d_in[7];
  const float* b1    = (const float*)d_in[8];
  const float* W2    = (const float*)d_in[9];
  const float* b2    = (const float*)d_in[10];
  const float* gamma = (const float*)d_in[11];
  const float* beta  = (const float*)d_in[12];
  float* out = (float*)d_out;

  float* gl    = (float*)d_ws;                    // [N,64]
  float* gr    = gl + (size_t)NNODES * 64;        // [E,64]
  float* score = gr + (size_t)NEDGES * 64;        // [E]
  float* agg   = score + (size_t)NEDGES;          // [N,64]

  proj64_kernel<<<NNODES / 16, 128, 0, stream>>>(h, Wl, gl);
  proj_edge_score_kernel<<<NEDGES / 16, 128, 0, stream>>>(e, Wr, recv, wa, gl,
                                                          gr, score);
  seg_softmax_agg_kernel<<<NNODES / 8, 256, 0, stream>>>(score, gr, recv, agg);
  mlp_ln_kernel<<<NNODES / 16, 128, 0, stream>>>(gl, agg, u, W1, b1, W2, b2,
                                                 gamma, beta, out);
}